// IDKN_7722351198324
// MI455X (gfx1250) — compile-verified
//
#include <hip/hip_runtime.h>
#include <hip/hip_bf16.h>

typedef float v2f __attribute__((ext_vector_type(2)));
typedef float v8f __attribute__((ext_vector_type(8)));

#define NEG_SLOPE 0.2f

__device__ __forceinline__ float leaky(float x) { return x > 0.0f ? x : NEG_SLOPE * x; }

// ---------------------------------------------------------------------------
// h0 = relu(x1 @ lin1_w^T + lin1_b)   x1:[N,7]
// ---------------------------------------------------------------------------
__global__ void k_h0(const float* __restrict__ x1, const float* __restrict__ w,
                     const float* __restrict__ b, float* __restrict__ h0, int N) {
  int n = blockIdx.x * blockDim.x + threadIdx.x;
  if (n >= N) return;
  float acc = b[0];
#pragma unroll
  for (int j = 0; j < 7; ++j) acc += x1[n * 7 + j] * w[j];
  h0[n] = fmaxf(acc, 0.0f);
}

// ---------------------------------------------------------------------------
// Layer-1 prep: h = h0 * W1 (outer product, Fin=1, Fout=32), attention dots,
// zero accumulators.
// ---------------------------------------------------------------------------
__global__ void k_l1_prep(const float* __restrict__ h0, const float* __restrict__ W1,
                          const float* __restrict__ a_src, const float* __restrict__ a_dst,
                          float* __restrict__ hfeat, float* __restrict__ als,
                          float* __restrict__ ald, float* __restrict__ num,
                          float* __restrict__ den, int N) {
  int n = blockIdx.x * blockDim.x + threadIdx.x;
  if (n >= N) return;
  float hv = h0[n];
#pragma unroll
  for (int j = 0; j < 32; ++j) {
    float hj = hv * W1[j];
    hfeat[(size_t)n * 32 + j] = hj;
    num[(size_t)n * 32 + j] = 0.0f;
  }
#pragma unroll
  for (int h = 0; h < 8; ++h) {
    float s = 0.0f, d = 0.0f;
#pragma unroll
    for (int c = 0; c < 4; ++c) {
      float hj = hv * W1[h * 4 + c];
      s += hj * a_src[h * 4 + c];
      d += hj * a_dst[h * 4 + c];
    }
    als[(size_t)n * 8 + h] = s;
    ald[(size_t)n * 8 + h] = d;
    den[(size_t)n * 8 + h] = 0.0f;
  }
}

// ---------------------------------------------------------------------------
// WMMA fp32 GEMM: Y[N,NOUT] = X[N,K] @ W[K,NOUT], K in {16,32}, NOUT<=16.
// One wave computes a 16-row M tile via a fully unrolled chain of
// V_WMMA_F32_16X16X4_F32. B fragments (wave-invariant, <=2KB) are hoisted out
// of the K chain; NOUT padding is branchless (clamped index * 0/1 mask) so
// EXEC stays all-ones through the WMMA chain as the ISA requires.
//   A layout: lane<16 holds M=lane, K=4kk+{0,1}; lane>=16: M=lane-16, K=4kk+{2,3}
//   B layout mirrors A; C/D: vgpr v -> M = v + (lane>=16 ? 8:0), N = lane&15
// ---------------------------------------------------------------------------
template <int K, int NOUT>
__global__ void __launch_bounds__(256) k_gemm_wmma(const float* __restrict__ X,
                                                   const float* __restrict__ W,
                                                   float* __restrict__ Y, int Nrows) {
  constexpr int KSTEPS = K / 4;
  int wave = threadIdx.x >> 5;
  int lane = threadIdx.x & 31;
  int rowBase = (blockIdx.x * (blockDim.x >> 5) + wave) * 16;
  if (rowBase >= Nrows) return;  // wave-uniform exit: EXEC all-ones below
  int m = lane & 15;
  int n = lane & 15;
  int kh = (lane >> 4) << 1;  // 0 or 2
  int r = rowBase + m;
  if (r >= Nrows) r = Nrows - 1;  // clamp loads; stores are guarded

  // Hoist B fragments (branchless column padding)
  int col = (n < NOUT) ? n : 0;
  float bmask = (n < NOUT) ? 1.0f : 0.0f;
  v2f bfrag[KSTEPS];
#pragma unroll
  for (int kk = 0; kk < KSTEPS; ++kk) {
    int k0 = 4 * kk + kh;
    bfrag[kk].x = bmask * W[k0 * NOUT + col];
    bfrag[kk].y = bmask * W[(k0 + 1) * NOUT + col];
  }

  v8f acc = {};
#pragma unroll
  for (int kk = 0; kk < KSTEPS; ++kk) {
    int k0 = 4 * kk + kh;
    v2f a;
    a.x = X[(size_t)r * K + k0];
    a.y = X[(size_t)r * K + k0 + 1];
    acc = __builtin_amdgcn_wmma_f32_16x16x4_f32(false, a, false, bfrag[kk], (short)0,
                                                acc, false, false);
  }

  int mo = (lane >> 4) * 8;
#pragma unroll
  for (int v = 0; v < 8; ++v) {
    int row = rowBase + mo + v;
    if (row < Nrows && n < NOUT) Y[(size_t)row * NOUT + n] = acc[v];
  }
}

// ---------------------------------------------------------------------------
// Generic per-node prep for layers 2/3: attention dots + zero accumulators.
// ---------------------------------------------------------------------------
template <int H, int C>
__global__ void k_prep(const float* __restrict__ hfeat, const float* __restrict__ a_src,
                       const float* __restrict__ a_dst, float* __restrict__ als,
                       float* __restrict__ ald, float* __restrict__ num,
                       float* __restrict__ den, int N) {
  constexpr int F = H * C;
  int n = blockIdx.x * blockDim.x + threadIdx.x;
  if (n >= N) return;
#pragma unroll
  for (int h = 0; h < H; ++h) {
    float s = 0.0f, d = 0.0f;
#pragma unroll
    for (int c = 0; c < C; ++c) {
      float hj = hfeat[(size_t)n * F + h * C + c];
      s += hj * a_src[h * C + c];
      d += hj * a_dst[h * C + c];
    }
    als[(size_t)n * H + h] = s;
    ald[(size_t)n * H + h] = d;
    den[(size_t)n * H + h] = 0.0f;
  }
#pragma unroll
  for (int j = 0; j < F; ++j) num[(size_t)n * F + j] = 0.0f;
}

// ---------------------------------------------------------------------------
// Edge scatter pass. Softmax max-shift skipped: scores are O(1), exp(e) is
// mathematically identical to the reference's exp(e - m) normalization.
// ---------------------------------------------------------------------------
template <int H, int C>
__global__ void k_edge(const int* __restrict__ src, const int* __restrict__ dst,
                       long long E, const float* __restrict__ als,
                       const float* __restrict__ ald, const float* __restrict__ hfeat,
                       float* __restrict__ num, float* __restrict__ den) {
  constexpr int F = H * C;
  long long e = (long long)blockIdx.x * blockDim.x + threadIdx.x;
  if (e >= E) return;
  if (e + 8192 < E) __builtin_prefetch(&src[e + 8192], 0, 1);  // global_prefetch_b8
  int s = src[e], d = dst[e];
  if (s == d) return;  // input self-loops are masked out
  float w[H];
#pragma unroll
  for (int h = 0; h < H; ++h) {
    float x = leaky(als[(size_t)s * H + h] + ald[(size_t)d * H + h]);
    w[h] = expf(x);
    atomicAdd(&den[(size_t)d * H + h], w[h]);
  }
#pragma unroll
  for (int h = 0; h < H; ++h)
#pragma unroll
    for (int c = 0; c < C; ++c)
      atomicAdd(&num[(size_t)d * F + h * C + c],
                w[h] * hfeat[(size_t)s * F + h * C + c]);
}

// ---------------------------------------------------------------------------
// Node finalize: add self loop, normalize, bias, relu. ZF=true also zeroes the
// final-score accumulators (used after layer 3).
// ---------------------------------------------------------------------------
template <int H, int C, bool ZF>
__global__ void k_finalize(const float* __restrict__ hfeat, const float* __restrict__ als,
                           const float* __restrict__ ald, const float* __restrict__ num,
                           const float* __restrict__ den, const float* __restrict__ bias,
                           float* __restrict__ xout, int N, float* __restrict__ flocal,
                           float* __restrict__ fdeg, float* __restrict__ fR3) {
  constexpr int F = H * C;
  int n = blockIdx.x * blockDim.x + threadIdx.x;
  if (n >= N) return;
#pragma unroll
  for (int h = 0; h < H; ++h) {
    float es = leaky(als[(size_t)n * H + h] + ald[(size_t)n * H + h]);
    float ws = expf(es);
    float dn = den[(size_t)n * H + h] + ws;
#pragma unroll
    for (int c = 0; c < C; ++c) {
      int j = h * C + c;
      float v = (num[(size_t)n * F + j] + ws * hfeat[(size_t)n * F + j]) / dn + bias[j];
      xout[(size_t)n * F + j] = fmaxf(v, 0.0f);
    }
  }
  if (ZF) {
    flocal[n] = 0.0f;
    fdeg[n] = 0.0f;
#pragma unroll
    for (int c = 0; c < 8; ++c) fR3[(size_t)n * 8 + c] = 0.0f;
  }
}

// ---------------------------------------------------------------------------
// Final edge pass: dots + R3 + degree, scattered over src.
// ---------------------------------------------------------------------------
__global__ void k_edge_final(const int* __restrict__ src, const int* __restrict__ dst,
                             long long E, const float* __restrict__ x6,
                             float* __restrict__ flocal, float* __restrict__ fdeg,
                             float* __restrict__ fR3) {
  long long e = (long long)blockIdx.x * blockDim.x + threadIdx.x;
  if (e >= E) return;
  if (e + 8192 < E) __builtin_prefetch(&src[e + 8192], 0, 1);
  int s = src[e], d = dst[e];
  if (s == d) return;
  float dot = 0.0f, xd[8];
#pragma unroll
  for (int c = 0; c < 8; ++c) {
    xd[c] = x6[(size_t)d * 8 + c];
    dot += x6[(size_t)s * 8 + c] * xd[c];
  }
  atomicAdd(&flocal[s], dot);
  atomicAdd(&fdeg[s], 1.0f);
#pragma unroll
  for (int c = 0; c < 8; ++c) atomicAdd(&fR3[(size_t)s * 8 + c], xd[c]);
}

__global__ void k_out(const float* __restrict__ x6, const float* __restrict__ flocal,
                      const float* __restrict__ fdeg, const float* __restrict__ fR3,
                      const float* __restrict__ lin2_w, float* __restrict__ out, int N) {
  int n = blockIdx.x * blockDim.x + threadIdx.x;
  if (n >= N) return;
  float deg = 1.0f + fdeg[n];
  float selfdot = 0.0f, g = 0.0f;
#pragma unroll
  for (int c = 0; c < 8; ++c) {
    float xv = x6[(size_t)n * 8 + c];
    selfdot += xv * xv;
    g += ((fR3[(size_t)n * 8 + c] + xv) / deg) * lin2_w[c];
  }
  out[n] = (flocal[n] + selfdot) / deg + g;
}

// ---------------------------------------------------------------------------
extern "C" void kernel_launch(void* const* d_in, const int* in_sizes, int n_in,
                              void* d_out, int out_size, void* d_ws, size_t ws_size,
                              hipStream_t stream) {
  const float* x1     = (const float*)d_in[0];
  const int*   ei     = (const int*)d_in[2];
  const float* lin1_w = (const float*)d_in[4];
  const float* lin1_b = (const float*)d_in[5];
  const float* lin2_w = (const float*)d_in[6];
  const float* W1     = (const float*)d_in[7];
  const float* a_src1 = (const float*)d_in[8];
  const float* a_dst1 = (const float*)d_in[9];
  const float* b1     = (const float*)d_in[10];
  const float* W2     = (const float*)d_in[11];
  const float* a_src2 = (const float*)d_in[12];
  const float* a_dst2 = (const float*)d_in[13];
  const float* b2     = (const float*)d_in[14];
  const float* W3     = (const float*)d_in[15];
  const float* a_src3 = (const float*)d_in[16];
  const float* a_dst3 = (const float*)d_in[17];
  const float* b3     = (const float*)d_in[18];

  const int N = in_sizes[0] / 7;                   // 100000
  const long long E = (long long)in_sizes[2] / 2;  // 3200000
  const int* srcp = ei;
  const int* dstp = ei + E;

  // Workspace layout (floats): 152 per node total (~61 MB)
  float* ws = (float*)d_ws;
  size_t Nn = (size_t)N;
  float* bufA   = ws;             // N*32  feature ping
  float* bufB   = ws + Nn * 32;   // N*32  feature pong
  float* bufH   = ws + Nn * 64;   // N*32  transformed features h
  float* bufNum = ws + Nn * 96;   // N*32  weighted sums / final accumulators
  float* bufALS = ws + Nn * 128;  // N*8
  float* bufALD = ws + Nn * 136;  // N*8
  float* bufDen = ws + Nn * 144;  // N*8

  const int TB = 256;
  dim3 blk(TB);
  dim3 gN((N + TB - 1) / TB);
  dim3 gE((unsigned)((E + TB - 1) / TB));
  dim3 gG((N + 127) / 128);  // gemm: 8 waves * 16 rows per block

  // ---- init score ----
  k_h0<<<gN, blk, 0, stream>>>(x1, lin1_w, lin1_b, bufA, N);

  // ---- GAT layer 1: h0[N,1] -> x3[N,32], H=8,C=4 ----
  k_l1_prep<<<gN, blk, 0, stream>>>(bufA, W1, a_src1, a_dst1, bufH, bufALS, bufALD,
                                    bufNum, bufDen, N);
  k_edge<8, 4><<<gE, blk, 0, stream>>>(srcp, dstp, E, bufALS, bufALD, bufH, bufNum,
                                       bufDen);
  k_finalize<8, 4, false><<<gN, blk, 0, stream>>>(bufH, bufALS, bufALD, bufNum, bufDen,
                                                  b1, bufB, N, nullptr, nullptr, nullptr);

  // ---- GAT layer 2: x3[N,32] @ W2[32,16], H=4,C=4 ----
  k_gemm_wmma<32, 16><<<gG, blk, 0, stream>>>(bufB, W2, bufH, N);
  k_prep<4, 4><<<gN, blk, 0, stream>>>(bufH, a_src2, a_dst2, bufALS, bufALD, bufNum,
                                       bufDen, N);
  k_edge<4, 4><<<gE, blk, 0, stream>>>(srcp, dstp, E, bufALS, bufALD, bufH, bufNum,
                                       bufDen);
  k_finalize<4, 4, false><<<gN, blk, 0, stream>>>(bufH, bufALS, bufALD, bufNum, bufDen,
                                                  b2, bufA, N, nullptr, nullptr, nullptr);

  // ---- GAT layer 3: x4[N,16] @ W3[16,8], H=2,C=4 ----
  k_gemm_wmma<16, 8><<<gG, blk, 0, stream>>>(bufA, W3, bufH, N);
  k_prep<2, 4><<<gN, blk, 0, stream>>>(bufH, a_src3, a_dst3, bufALS, bufALD, bufNum,
                                       bufDen, N);
  k_edge<2, 4><<<gE, blk, 0, stream>>>(srcp, dstp, E, bufALS, bufALD, bufH, bufNum,
                                       bufDen);
  // x6 -> bufB; also zero final accumulators living in bufNum
  float* flocal = bufNum;
  float* fdeg   = bufNum + Nn;
  float* fR3    = bufNum + 2 * Nn;
  k_finalize<2, 4, true><<<gN, blk, 0, stream>>>(bufH, bufALS, bufALD, bufNum, bufDen,
                                                 b3, bufB, N, flocal, fdeg, fR3);

  // ---- final local/global score ----
  k_edge_final<<<gE, blk, 0, stream>>>(srcp, dstp, E, bufB, flocal, fdeg, fR3);
  k_out<<<gN, blk, 0, stream>>>(bufB, flocal, fdeg, fR3, lin2_w, (float*)d_out, N);
}